// PairwiseProjector_10582799417763
// MI455X (gfx1250) — compile-verified
//
#include <hip/hip_runtime.h>
#include <math.h>

typedef __attribute__((ext_vector_type(2))) float v2f;
typedef __attribute__((ext_vector_type(8))) float v8f;

#define LL 256
#define DD 1280
#define OCH 128
#define SF 1284   /* feats LDS row stride: 1284 % 64 == 4 -> conflict-free b64 across 16 rows */
#define WST 36    /* W-chunk LDS row stride: 36*nl distinct mod 64, rows 16B aligned */
#define NCH 40    /* K-chunks per section: 1280 / 32 */

// ---------------- K1: A[i][o] = feats[i,:] . W[o, 0:D]  (add-term decomposition) ----------------
__global__ void k_addproj(const float* __restrict__ feats,
                          const float* __restrict__ W,
                          float* __restrict__ A) {
  __shared__ float row[DD];
  const int i = blockIdx.x;
  for (int c = threadIdx.x; c < DD; c += 128) row[c] = feats[i * DD + c];
  __syncthreads();
  const int o = threadIdx.x;
  const float* wr = W + (size_t)o * (3 * DD);
  float s = 0.f;
#pragma unroll 4
  for (int c = 0; c < DD; c += 4) {
    s += row[c] * wr[c];
    s += row[c + 1] * wr[c + 1];
    s += row[c + 2] * wr[c + 2];
    s += row[c + 3] * wr[c + 3];
  }
  A[i * OCH + o] = s;
}

// ---------------- K2: WMMA pair-GEMM, waves = 4 pair-groups x 2 channel-halves ----------------
__global__ void __launch_bounds__(256, 1)
k_pair_gemm(const float* __restrict__ feats, const float* __restrict__ W,
            const float* __restrict__ A, const float* __restrict__ bias,
            float* __restrict__ xbuf, float* __restrict__ parts) {
  __shared__ float sFi[16 * SF];
  __shared__ float sFj[16 * SF];
  __shared__ float sW[2][128 * WST];
  __shared__ float sAi[16 * OCH];
  __shared__ float sAj[16 * OCH];
  __shared__ float sB[OCH];

  const int tid = threadIdx.x;
  const int wg = blockIdx.x;
  const int i0 = (wg >> 4) << 4;
  const int j0 = (wg & 15) << 4;

  // Stage both 16x1280 f32 feature tiles (row-padded), A tiles, bias.
  for (int t = tid; t < 16 * (DD / 4); t += 256) {
    const int r = t / (DD / 4);
    const int c4 = t - r * (DD / 4);
    ((float4*)(sFi + r * SF))[c4] = ((const float4*)(feats + (size_t)(i0 + r) * DD))[c4];
    ((float4*)(sFj + r * SF))[c4] = ((const float4*)(feats + (size_t)(j0 + r) * DD))[c4];
  }
  for (int t = tid; t < 16 * OCH / 4; t += 256) {
    ((float4*)sAi)[t] = ((const float4*)(A + i0 * OCH))[t];
    ((float4*)sAj)[t] = ((const float4*)(A + j0 * OCH))[t];
  }
  if (tid < OCH) sB[tid] = bias[tid];

  // Stage W chunk 0 of |diff| section (cols [DD, DD+32)).
#pragma unroll
  for (int r = 0; r < 4; ++r) {
    const int u = tid + (r << 8);
    const int o = u >> 3, k4 = (u & 7) << 2;
    *(float4*)(&sW[0][o * WST + k4]) = *(const float4*)(W + (size_t)o * (3 * DD) + DD + k4);
  }
  __syncthreads();

  const int wave = tid >> 5;
  const int lane = tid & 31;
  const int half = lane >> 4;       // K half (A/B frags) and M half (C/D frags)
  const int nl = lane & 15;         // A row (j) / B,C,D column (o)
  const int kk = half << 1;         // lane K offset within 4-wide K step
  const int pg = wave & 3;          // pair group: i rows 4*pg .. 4*pg+3
  const int cg = wave >> 2;         // channel half: channels 64*cg .. 64*cg+63

  const float* fjp = sFj + nl * SF + kk;
  const float* fip = sFi + (pg << 2) * SF + kk;
  const int wrow = ((cg << 6) + nl) * WST + kk;  // lane's nt=0 row offset into sW chunk

  v8f acc[4][4];  // [m][nt]
#pragma unroll
  for (int m = 0; m < 4; ++m)
#pragma unroll
    for (int nt = 0; nt < 4; ++nt) acc[m][nt] = (v8f)0.f;

  // ================= |fi - fj| section =================
  for (int ch = 0; ch < NCH; ++ch) {
    const int buf = ch & 1;
    const bool more = (ch + 1) < NCH;
    float4 pre[4];
    if (more) {
      const int c0 = DD + ((ch + 1) << 5);
#pragma unroll
      for (int r = 0; r < 4; ++r) {
        const int u = tid + (r << 8);
        const int o = u >> 3, k4 = (u & 7) << 2;
        pre[r] = *(const float4*)(W + (size_t)o * (3 * DD) + c0 + k4);
      }
    }
    const float* Wb = sW[buf];
#pragma unroll 2
    for (int kl = 0; kl < 8; ++kl) {
      const int cl = kl << 2;
      const int cc = (ch << 5) + cl;
      const float2 fj2 = *(const float2*)(fjp + cc);
      v2f bfr[4];
#pragma unroll
      for (int nt = 0; nt < 4; ++nt) {
        const float2 t2 = *(const float2*)(Wb + wrow + nt * (16 * WST) + cl);
        bfr[nt].x = t2.x;
        bfr[nt].y = t2.y;
      }
#pragma unroll
      for (int m = 0; m < 4; ++m) {
        const float2 fi2 = *(const float2*)(fip + m * SF + cc);
        v2f a;
        a.x = fabsf(fi2.x - fj2.x);
        a.y = fabsf(fi2.y - fj2.y);
#pragma unroll
        for (int nt = 0; nt < 4; ++nt)
          acc[m][nt] = __builtin_amdgcn_wmma_f32_16x16x4_f32(false, a, false, bfr[nt],
                                                             (short)0, acc[m][nt], false, false);
      }
    }
    if (more) {
#pragma unroll
      for (int r = 0; r < 4; ++r) {
        const int u = tid + (r << 8);
        const int o = u >> 3, k4 = (u & 7) << 2;
        *(float4*)(&sW[buf ^ 1][o * WST + k4]) = pre[r];
      }
    }
    __syncthreads();
  }

  // Stage W chunk 0 of hadamard section (cols [2*DD, 2*DD+32)).
#pragma unroll
  for (int r = 0; r < 4; ++r) {
    const int u = tid + (r << 8);
    const int o = u >> 3, k4 = (u & 7) << 2;
    *(float4*)(&sW[0][o * WST + k4]) = *(const float4*)(W + (size_t)o * (3 * DD) + 2 * DD + k4);
  }
  __syncthreads();

  // ================= fi * fj section =================
  for (int ch = 0; ch < NCH; ++ch) {
    const int buf = ch & 1;
    const bool more = (ch + 1) < NCH;
    float4 pre[4];
    if (more) {
      const int c0 = 2 * DD + ((ch + 1) << 5);
#pragma unroll
      for (int r = 0; r < 4; ++r) {
        const int u = tid + (r << 8);
        const int o = u >> 3, k4 = (u & 7) << 2;
        pre[r] = *(const float4*)(W + (size_t)o * (3 * DD) + c0 + k4);
      }
    }
    const float* Wb = sW[buf];
#pragma unroll 2
    for (int kl = 0; kl < 8; ++kl) {
      const int cl = kl << 2;
      const int cc = (ch << 5) + cl;
      const float2 fj2 = *(const float2*)(fjp + cc);
      v2f bfr[4];
#pragma unroll
      for (int nt = 0; nt < 4; ++nt) {
        const float2 t2 = *(const float2*)(Wb + wrow + nt * (16 * WST) + cl);
        bfr[nt].x = t2.x;
        bfr[nt].y = t2.y;
      }
#pragma unroll
      for (int m = 0; m < 4; ++m) {
        const float2 fi2 = *(const float2*)(fip + m * SF + cc);
        v2f a;
        a.x = fi2.x * fj2.x;
        a.y = fi2.y * fj2.y;
#pragma unroll
        for (int nt = 0; nt < 4; ++nt)
          acc[m][nt] = __builtin_amdgcn_wmma_f32_16x16x4_f32(false, a, false, bfr[nt],
                                                             (short)0, acc[m][nt], false, false);
      }
    }
    if (more) {
#pragma unroll
      for (int r = 0; r < 4; ++r) {
        const int u = tid + (r << 8);
        const int o = u >> 3, k4 = (u & 7) << 2;
        *(float4*)(&sW[buf ^ 1][o * WST + k4]) = pre[r];
      }
    }
    __syncthreads();
  }

  // ---- epilogue: + A[i,o] + A[j,o] + b[o]; per-group stats; store x[o][i][j] ----
#pragma unroll
  for (int nt = 0; nt < 4; ++nt) {
    const int oc = (cg << 6) + (nt << 4) + nl;
    const float bb = sB[oc];
    float ajv[8];
#pragma unroll
    for (int v = 0; v < 8; ++v) ajv[v] = sAj[(v + (half << 3)) * OCH + oc] + bb;
    float s = 0.f, q = 0.f;
#pragma unroll
    for (int m = 0; m < 4; ++m) {
      const int im = (pg << 2) + m;
      const float ai = sAi[im * OCH + oc];
#pragma unroll
      for (int v = 0; v < 8; ++v) {
        float x = acc[m][nt][v] + ai + ajv[v];
        acc[m][nt][v] = x;
        s += x;
        q += x * x;
      }
      float* p = xbuf + (size_t)oc * (LL * LL) + (size_t)(i0 + im) * LL + j0 + (half << 3);
      float4 lo = make_float4(acc[m][nt][0], acc[m][nt][1], acc[m][nt][2], acc[m][nt][3]);
      float4 hi = make_float4(acc[m][nt][4], acc[m][nt][5], acc[m][nt][6], acc[m][nt][7]);
      ((float4*)p)[0] = lo;
      ((float4*)p)[1] = hi;
    }
#pragma unroll
    for (int off = 16; off > 0; off >>= 1) {
      s += __shfl_xor(s, off, 32);
      q += __shfl_xor(q, off, 32);
    }
    if (lane == 0) {
      const int g = (cg << 2) + nt;
      float* pp = parts + ((((size_t)wg << 3) + g) * 4 + pg) * 2;
      pp[0] = s;
      pp[1] = q;
    }
  }
}

// ---------------- K3: reduce 1024 partials per group -> mean, rstd ----------------
__global__ void k_stats(const float* __restrict__ parts, float* __restrict__ stats) {
  const int g = threadIdx.x >> 5;  // 8 groups, one wave each
  const int lane = threadIdx.x & 31;
  float s = 0.f, q = 0.f;
  for (int e = lane; e < 1024; e += 32) {
    const int w = e >> 2, pgi = e & 3;
    const int p = ((w * 8 + g) * 4 + pgi) * 2;
    s += parts[p];
    q += parts[p + 1];
  }
#pragma unroll
  for (int off = 16; off > 0; off >>= 1) {
    s += __shfl_xor(s, off, 32);
    q += __shfl_xor(q, off, 32);
  }
  if (lane == 0) {
    const float n = 16.f * 65536.f;  // 16 channels * 256*256
    const float mean = s / n;
    const float var = q / n - mean * mean;
    stats[g * 2] = mean;
    stats[g * 2 + 1] = rsqrtf(var + 1e-5f);
  }
}

// ---------------- K4: streaming normalize + exact GELU ----------------
__global__ void k_norm_gelu(const float* __restrict__ xbuf, const float* __restrict__ stats,
                            const float* __restrict__ gamma, const float* __restrict__ beta,
                            float* __restrict__ out) {
  const int idx = blockIdx.x * 256 + threadIdx.x;  // float4 index
  const int o = idx >> 14;                          // 65536/4 float4 per channel
  const int g = o >> 4;
  const float mean = stats[g * 2];
  const float rstd = stats[g * 2 + 1];
  const float sc = gamma[o] * rstd;
  const float sh = beta[o] - mean * sc;
  const float4 x = ((const float4*)xbuf)[idx];
  float4 r;
  float y;
  y = x.x * sc + sh; r.x = 0.5f * y * (1.f + erff(y * 0.70710678118654752f));
  y = x.y * sc + sh; r.y = 0.5f * y * (1.f + erff(y * 0.70710678118654752f));
  y = x.z * sc + sh; r.z = 0.5f * y * (1.f + erff(y * 0.70710678118654752f));
  y = x.w * sc + sh; r.w = 0.5f * y * (1.f + erff(y * 0.70710678118654752f));
  ((float4*)out)[idx] = r;
}

extern "C" void kernel_launch(void* const* d_in, const int* in_sizes, int n_in,
                              void* d_out, int out_size, void* d_ws, size_t ws_size,
                              hipStream_t stream) {
  const float* feats = (const float*)d_in[0];
  const float* W = (const float*)d_in[1];
  const float* b = (const float*)d_in[2];
  const float* gamma = (const float*)d_in[3];
  const float* beta = (const float*)d_in[4];
  float* out = (float*)d_out;

  // workspace: x[128][256][256] | A[256][128] | partials[256][8][4][2] | stats[8][2]
  float* xbuf = (float*)d_ws;
  float* A = xbuf + (size_t)OCH * LL * LL;
  float* parts = A + (size_t)LL * OCH;
  float* stats = parts + 256 * 8 * 4 * 2;

  k_addproj<<<LL, 128, 0, stream>>>(feats, W, A);
  k_pair_gemm<<<256, 256, 0, stream>>>(feats, W, A, b, xbuf, parts);
  k_stats<<<1, 256, 0, stream>>>(parts, stats);
  k_norm_gelu<<<(OCH * LL * LL / 4) / 256, 256, 0, stream>>>(xbuf, stats, gamma, beta, out);
}